// PSD_11587821765058
// MI455X (gfx1250) — compile-verified
//
#include <hip/hip_runtime.h>

typedef float v2f __attribute__((ext_vector_type(2)));
typedef float v8f __attribute__((ext_vector_type(8)));

#define B_  8
#define F_  257
#define C_  8
#define T_  3000
#define EPSF 1e-15f

// One wave (32 lanes) handles one (b,f) bin:
//   D(16x16) = (mask-weighted [Xr;Xi]) x [Xr;Xi]^T  via V_WMMA_F32_16X16X4_F32
// then complex recombination:
//   psd_re[c][e] = D[c][e] + D[c+8][e+8]
//   psd_im[c][e] = D[c+8][e] - D[c][e+8]
__global__ __launch_bounds__(256) void psd_wmma_f32_kernel(
    const float* __restrict__ Xr,
    const float* __restrict__ Xi,
    const float* __restrict__ mask,
    float* __restrict__ out)
{
    const int lane = threadIdx.x & 31;
    const int wave = threadIdx.x >> 5;
    const int bf   = blockIdx.x * 8 + wave;      // 257 blocks * 8 waves = 2056 = B*F exactly

    const int hv = lane >> 4;                    // which K-half this lane supplies (k 0-1 vs 2-3)
    const int j  = lane & 15;                    // A-row / B-col index 0..15

    // ---- mask normalization: sum over T, then fold 1/(sum+eps) into A ----
    const float* mrow = mask + (size_t)bf * T_;
    float s = 0.f;
    for (int t = lane; t < T_; t += 32) s += mrow[t];
    #pragma unroll
    for (int off = 16; off > 0; off >>= 1) s += __shfl_xor(s, off, 32);
    const float inv = 1.0f / (s + EPSF);

    // stacked matrix S(16 x T): rows 0-7 = X_real, rows 8-15 = X_imag
    const float* row = (j < 8)
        ? (Xr + ((size_t)bf * C_ + j) * T_)
        : (Xi + ((size_t)bf * C_ + (j - 8)) * T_);

    v8f acc = {};

    // K-loop: 8 t-values per iteration -> one float4 per lane, two WMMAs.
    // Lane layout for V_WMMA_F32_16X16X4_F32 (32-bit A 16x4 / B 4x16):
    //   lane(h,j): A slots {k=2h, k=2h+1} of row j; B slots {k=2h, k=2h+1} of col j.
    // A and B read the SAME addresses (A is mask-weighted), so one load feeds both.
    for (int t0 = 0; t0 < T_; t0 += 8) {
        const int tl = t0 + 4 * hv;                       // 16B-aligned (row offsets are 16B multiples)
        const float4 xv = *(const float4*)(row  + tl);
        const float4 mv = *(const float4*)(mrow + tl);    // broadcast across the 16 lanes of each half

        v2f a0; a0.x = xv.x * (mv.x * inv); a0.y = xv.y * (mv.y * inv);
        v2f b0; b0.x = xv.x;                b0.y = xv.y;
        acc = __builtin_amdgcn_wmma_f32_16x16x4_f32(false, a0, false, b0,
                                                    (short)0, acc, false, false);

        v2f a1; a1.x = xv.z * (mv.z * inv); a1.y = xv.w * (mv.w * inv);
        v2f b1; b1.x = xv.z;                b1.y = xv.w;
        acc = __builtin_amdgcn_wmma_f32_16x16x4_f32(false, a1, false, b1,
                                                    (short)0, acc, false, false);
    }

    // ---- stage D fragment to LDS, recombine into complex psd ----
    // C/D layout: VGPR r of lane(h,j) holds D[r + 8h][j]
    __shared__ float Dl[8][16][17];
    #pragma unroll
    for (int r = 0; r < 8; ++r)
        Dl[wave][r + 8 * hv][j] = acc[r];
    __syncthreads();

    // 64 complex outputs per bin, 2 per lane; interleaved (re, im)
    float* obase = out + (size_t)bf * (C_ * C_ * 2);
    #pragma unroll
    for (int q = 0; q < 2; ++q) {
        const int idx = lane * 2 + q;       // 0..63
        const int c = idx >> 3, e = idx & 7;
        const float re = Dl[wave][c][e]     + Dl[wave][c + 8][e + 8];
        const float im = Dl[wave][c + 8][e] - Dl[wave][c][e + 8];
        obase[idx * 2 + 0] = re;
        obase[idx * 2 + 1] = im;
    }
}

extern "C" void kernel_launch(void* const* d_in, const int* in_sizes, int n_in,
                              void* d_out, int out_size, void* d_ws, size_t ws_size,
                              hipStream_t stream) {
    const float* Xr   = (const float*)d_in[0];
    const float* Xi   = (const float*)d_in[1];
    const float* mask = (const float*)d_in[2];
    float* out = (float*)d_out;

    dim3 grid((B_ * F_) / 8);   // 257 blocks, 8 waves each -> one wave per (b,f)
    dim3 block(256);
    psd_wmma_f32_kernel<<<grid, block, 0, stream>>>(Xr, Xi, mask, out);
}